// SparseFCN_84482006712502
// MI455X (gfx1250) — compile-verified
//
#include <hip/hip_runtime.h>
#include <stdint.h>

typedef __attribute__((ext_vector_type(2))) float v2f;
typedef __attribute__((ext_vector_type(8))) float v8f;
typedef float v2f_u __attribute__((ext_vector_type(2), aligned(4)));  // 4B-aligned pair store

// MODE 0: forward conv, stride 2, dilation 2, VALID:
//         in(iy,ix) = (2*py + 2*ky, 2*px + 2*t).
// MODE 1: inverse (transposed) conv. Matching lhs/rhs dilation makes every
//         odd output coordinate pure bias; even-even outputs reduce to
//         out[2*py][2*px] = b + sum w[ky][kx] * in[py-ky][px-kx].
//         Taps enumerated with t = 2-kx so ix = px-2+t ascends with k
//         (adjacent k -> adjacent addresses -> b64-mergeable loads);
//         weights are re-permuted to match during LDS staging.
//         The epilogue writes the full 4-row x 32-col patch (computed even
//         columns + bias odd columns + bias odd rows) so no separate fill
//         pass is needed and every output byte is written exactly once.
//
// GEMM mapping per wave: M = 16 output-channel slots (COUT padded),
// N = 16 consecutive pixels, K = CIN*9 (constexpr, fully unrolled).
// Each wave produces TWO computed rows from one A-fragment stream.

template<int MODE, int CIN, int INH, int INW, int POSH, int POSW, bool CHECK>
__device__ __forceinline__ void conv_kloop(const float* __restrict__ inN,
                                           const float* __restrict__ wldsp,
                                           int px, int py0, int half, int l16,
                                           v8f& c0, v8f& c1)
{
    constexpr int  Kreal   = CIN * 9;
    constexpr int  Kpad    = (Kreal + 3) & ~3;
    constexpr long inPlane = (long)INH * INW;

    auto loadB = [&](int k, int py) -> float {
        if (k >= Kreal) return 0.f;                 // constant-folded
        const int ci = k / 9;                       // constexpr after unroll
        const int r  = k - ci * 9;
        const int ky = r / 3;
        const int t  = r - ky * 3;
        if (CHECK && (px >= POSW || py >= POSH)) return 0.f;
        int iy, ix;
        if (MODE == 0) { iy = 2 * py + 2 * ky; ix = 2 * px + 2 * t; }
        else {
            iy = py - ky; ix = px - 2 + t;
            if (CHECK && (iy < 0 || iy >= INH || ix < 0 || ix >= INW)) return 0.f;
        }
        return inN[ci * inPlane + (long)iy * INW + ix];
    };

    #pragma unroll
    for (int kb = 0; kb < Kpad; kb += 4) {
        const int ka = kb + 2 * half;   // ISA A layout: VGPR0=K+0/K+2, VGPR1=K+1/K+3
        v2f a = *(const v2f*)&wldsp[(ka >> 1) * 32 + l16 * 2];   // ds_load_b64
        v2f b0, b1;
        b0.x = loadB(ka,     py0);
        b0.y = loadB(ka + 1, py0);
        b1.x = loadB(ka,     py0 + 1);
        b1.y = loadB(ka + 1, py0 + 1);
        // 8 args: (neg_a, A, neg_b, B, c_mod, C, reuse_a, reuse_b)
        c0 = __builtin_amdgcn_wmma_f32_16x16x4_f32(
                 false, a, false, b0, (short)0, c0, false, false);
        c1 = __builtin_amdgcn_wmma_f32_16x16x4_f32(
                 false, a, false, b1, (short)0, c1, false, false);
    }
}

template<int MODE, int RELU, int CIN, int COUT,
         int INH, int INW, int POSH, int POSW, int OUTH, int OUTW, int NB>
__global__ void conv_wmma_kernel(const float* __restrict__ in,
                                 const float* __restrict__ wgt,
                                 const float* __restrict__ bias,
                                 float* __restrict__ out)
{
    constexpr int Kreal  = CIN * 9;
    constexpr int Kpad   = (Kreal + 3) & ~3;
    constexpr int tilesX = (POSW + 15) / 16;
    constexpr int rowsH  = (POSH + 1) / 2;
    constexpr int total  = NB * rowsH * tilesX;
    __shared__ float wlds[Kpad * 16];

    // Stage weights: pair-interleaved A-fragment layout
    // wlds[(k>>1)*32 + co*2 + (k&1)], with MODE-1 x-flip applied to the tap.
    for (int idx = threadIdx.x; idx < Kpad * 16; idx += blockDim.x) {
        const int k  = ((idx >> 5) << 1) | (idx & 1);
        const int co = (idx >> 1) & 15;
        float v = 0.f;
        if (k < Kreal && co < COUT) {
            const int ci = k / 9, r = k - ci * 9, ky = r / 3, t = r - ky * 3;
            const int srcK = ci * 9 + ky * 3 + (MODE == 1 ? 2 - t : t);
            v = wgt[co * Kreal + srcK];
        }
        wlds[idx] = v;
    }
    __syncthreads();

    const int wave = blockIdx.x * (blockDim.x >> 5) + (threadIdx.x >> 5);
    if (wave >= total) return;               // wave-uniform: EXEC stays all-1s

    const int lane = threadIdx.x & 31;
    const int half = lane >> 4;              // 0 = lanes 0-15, 1 = lanes 16-31
    const int l16  = lane & 15;

    const int tx  = wave % tilesX;
    const int py0 = ((wave / tilesX) % rowsH) * 2;
    const int py1 = py0 + 1;
    const int n   = wave / (tilesX * rowsH);
    const int px0 = tx * 16;
    const int px  = px0 + l16;               // this lane's pixel column (N role)

    const float* inN = in + (long)n * CIN * ((long)INH * INW);

    // Interior (memory-safety) test — wave-uniform.
    bool interior;
    if (MODE == 0)
        interior = (px0 + 16 <= POSW) && (py0 + 2 <= POSH);
    else
        interior = (px0 >= 2) && (px0 + 16 <= INW) &&
                   (py0 >= 2) && (py0 + 2 <= INH);

    v8f c0 = {}, c1 = {};
    if (interior)
        conv_kloop<MODE, CIN, INH, INW, POSH, POSW, false>(
            inN, wlds, px, py0, half, l16, c0, c1);
    else
        conv_kloop<MODE, CIN, INH, INW, POSH, POSW, true>(
            inN, wlds, px, py0, half, l16, c0, c1);

    // D layout: VGPR r -> M = r (lanes 0-15) / r+8 (lanes 16-31); N = lane&15.
    constexpr long outPlane = (long)OUTH * OUTW;

    if (MODE == 0) {
        if (px < POSW) {
            #pragma unroll
            for (int r = 0; r < 8; ++r) {
                const int co = r + 8 * half;
                if (co < COUT) {
                    const float b = bias[co];
                    float* op = out + ((long)n * COUT + co) * outPlane;
                    float v0 = c0[r] + b;
                    if (RELU) v0 = fmaxf(v0, 0.f);
                    op[(long)py0 * OUTW + px] = v0;
                    if (py1 < POSH) {
                        float v1 = c1[r] + b;
                        if (RELU) v1 = fmaxf(v1, 0.f);
                        op[(long)py1 * OUTW + px] = v1;
                    }
                }
            }
        }
    } else {
        // Own the full 4-row x (2 cols per lane) patch: rows 2*py0 .. 2*py0+3.
        const int ox = 2 * px;               // lane's even column
        #pragma unroll
        for (int r = 0; r < 8; ++r) {
            const int co = r + 8 * half;
            if (co < COUT) {
                const float b   = bias[co];
                const float bfv = RELU ? fmaxf(b, 0.f) : b;   // bias fill value
                float* op = out + ((long)n * COUT + co) * outPlane;

                auto storeRow = [&](int oy, float evenVal) {
                    if (oy >= OUTH) return;
                    float* q = op + (long)oy * OUTW + ox;
                    if (ox + 1 < OUTW) { v2f_u s = {evenVal, bfv}; *(v2f_u*)q = s; }
                    else if (ox < OUTW) { *q = evenVal; }
                };

                float v0 = c0[r] + b;
                float v1 = c1[r] + b;
                if (RELU) { v0 = fmaxf(v0, 0.f); v1 = fmaxf(v1, 0.f); }
                storeRow(2 * py0,     v0);    // computed even row
                storeRow(2 * py0 + 1, bfv);   // bias-only odd row
                storeRow(2 * py0 + 2, v1);    // computed even row (oy>=OUTH iff py1>=POSH)
                storeRow(2 * py0 + 3, bfv);   // bias-only odd row
            }
        }
    }
}

static inline int ceil_div(long a, long b) { return (int)((a + b - 1) / b); }

extern "C" void kernel_launch(void* const* d_in, const int* in_sizes, int n_in,
                              void* d_out, int out_size, void* d_ws, size_t ws_size,
                              hipStream_t stream)
{
    const float* x  = (const float*)d_in[0];
    const float* w1 = (const float*)d_in[1];
    const float* b1 = (const float*)d_in[2];
    const float* w2 = (const float*)d_in[3];
    const float* b2 = (const float*)d_in[4];
    const float* w3 = (const float*)d_in[5];
    const float* b3 = (const float*)d_in[6];
    const float* w4 = (const float*)d_in[7];
    const float* b4 = (const float*)d_in[8];
    float* out = (float*)d_out;

    constexpr int NB = 8;
    constexpr int H0 = 1021, H1 = 509, H2 = 253, H3 = 509, HO = 1021;

    // Workspace carve-up (bytes): h1 | h2 | h3
    char* ws = (char*)d_ws;
    float* h1 = (float*)ws;
    float* h2 = (float*)(ws + (((size_t)NB * 8  * H1 * H1 * 4 + 255) & ~(size_t)255));
    float* h3 = (float*)((char*)h2 + (((size_t)NB * 16 * H2 * H2 * 4 + 255) & ~(size_t)255));

    const int BLOCK = 256, WPB = BLOCK / 32;

    // ---- L1: fwd conv 3->8, 1021 -> 509, relu ----
    {
        int waves = NB * ((H1 + 1) / 2) * ceil_div(H1, 16);
        conv_wmma_kernel<0, 1, 3, 8, H0, H0, H1, H1, H1, H1, NB>
            <<<ceil_div(waves, WPB), BLOCK, 0, stream>>>(x, w1, b1, h1);
    }
    // ---- L2: fwd conv 8->16, 509 -> 253, relu ----
    {
        int waves = NB * ((H2 + 1) / 2) * ceil_div(H2, 16);
        conv_wmma_kernel<0, 1, 8, 16, H1, H1, H2, H2, H2, H2, NB>
            <<<ceil_div(waves, WPB), BLOCK, 0, stream>>>(h1, w2, b2, h2);
    }
    // ---- L3: inverse conv 16->8, 253 -> 509, relu (fused bias fill) ----
    {
        int waves = NB * ((255 + 1) / 2) * ceil_div(255, 16);
        conv_wmma_kernel<1, 1, 16, 8, H2, H2, 255, 255, H3, H3, NB>
            <<<ceil_div(waves, WPB), BLOCK, 0, stream>>>(h2, w3, b3, h3);
    }
    // ---- L4: inverse conv 8->8, 509 -> 1021, no relu (fused bias fill) ----
    {
        int waves = NB * ((511 + 1) / 2) * ceil_div(511, 16);
        conv_wmma_kernel<1, 0, 8, 8, H3, H3, 511, 511, HO, HO, NB>
            <<<ceil_div(waves, WPB), BLOCK, 0, stream>>>(h3, w4, b4, out);
    }
}